// SNGNN_Plus_62689342652828
// MI455X (gfx1250) — compile-verified
//
#include <hip/hip_runtime.h>
#include <hip/hip_bf16.h>

typedef float v2f __attribute__((ext_vector_type(2)));
typedef float v8f __attribute__((ext_vector_type(8)));

#define IN_DIM  128
#define HID_DIM 128
#define OUT_DIM 64
#define KDIM    128   // contraction dim for both layers

// ---------- order-preserving float key (bigger w -> bigger key) ----------
__device__ __forceinline__ unsigned fkey_enc(float w) {
  unsigned b = __float_as_uint(w);
  return (b & 0x80000000u) ? ~b : (b | 0x80000000u);
}
__device__ __forceinline__ float fkey_dec(unsigned k) {
  return __uint_as_float((k & 0x80000000u) ? (k & 0x7FFFFFFFu) : ~k);
}

// =====================================================================
// GEMM: Y[N x C] = X[N x 128] @ W[128 x C] + bias, via V_WMMA_F32_16X16X4_F32
// Block: C/16 waves; each wave owns one 16x16 output tile (same 16 rows,
// distinct 16-col slice). A-tile staged in LDS (padded rows -> conflict-free
// strided fragment reads). W stays in global: it is 32-64 KB, L2/L0 resident.
// =====================================================================
template <int C>
__global__ __launch_bounds__(256) void gemm_wmma_kernel(
    const float* __restrict__ X, const float* __restrict__ W,
    const float* __restrict__ bias, float* __restrict__ Y, int N) {
  __shared__ float As[16][KDIM + 2];  // +2 pad: fragment reads stride 130 -> distinct banks
  const int tid  = threadIdx.x;
  const int lane = tid & 31;
  const int wave = tid >> 5;
  const long m0  = (long)blockIdx.x * 16;

  // cooperative, coalesced load of the 16 x 128 A tile
  for (int i = tid; i < 16 * KDIM; i += blockDim.x) {
    int r = i >> 7, c = i & (KDIM - 1);
    As[r][c] = (m0 + r < N) ? X[(m0 + r) * KDIM + c] : 0.0f;
  }
  __syncthreads();

  const int n0  = wave * 16;
  const int l15 = lane & 15;          // A row within tile / B col within slice
  const int kb  = (lane >> 4) << 1;   // lanes 16-31 hold K+2,K+3 halves

  v8f acc;
#pragma unroll
  for (int i = 0; i < 8; ++i) acc[i] = 0.0f;

#pragma unroll 4
  for (int k0 = 0; k0 < KDIM; k0 += 4) {
    v2f a, b;
    a[0] = As[l15][k0 + kb];
    a[1] = As[l15][k0 + kb + 1];
    b[0] = W[(k0 + kb) * C + n0 + l15];
    b[1] = W[(k0 + kb + 1) * C + n0 + l15];
    acc = __builtin_amdgcn_wmma_f32_16x16x4_f32(
        /*neg_a=*/false, a, /*neg_b=*/false, b,
        /*c_mod=*/(short)0, acc, /*reuse_a=*/false, /*reuse_b=*/false);
  }

  // C/D layout: vgpr i holds row (i + 8*(lane>>4)), col = lane&15 (of slice)
  const int   ncol = n0 + l15;
  const float bv   = bias[ncol];
  const int   mhi  = (lane >> 4) << 3;
  if (m0 + 16 <= (long)N) {
    // fast path: whole tile in range -> one base, immediate offsets
    float* __restrict__ yp = Y + (size_t)(m0 + mhi) * C + ncol;
#pragma unroll
    for (int i = 0; i < 8; ++i) yp[(size_t)i * C] = acc[i] + bv;
  } else {
#pragma unroll
    for (int i = 0; i < 8; ++i) {
      long m = m0 + mhi + i;
      if (m < N) Y[m * C + ncol] = acc[i] + bv;
    }
  }
}

// ---------------- per-node inverse L2 norm (one wave per node) ----------------
template <int C>
__global__ __launch_bounds__(256) void rownorm_kernel(
    const float* __restrict__ H, float* __restrict__ invn, int N) {
  int wid = blockIdx.x * (blockDim.x >> 5) + (threadIdx.x >> 5);
  if (wid >= N) return;
  int lane = threadIdx.x & 31;
  float acc;
  if constexpr (C == 128) {
    float4 v = *(const float4*)(H + (size_t)wid * C + lane * 4);
    acc = v.x * v.x + v.y * v.y + v.z * v.z + v.w * v.w;
  } else {
    float2 v = *(const float2*)(H + (size_t)wid * C + lane * 2);
    acc = v.x * v.x + v.y * v.y;
  }
  for (int off = 16; off; off >>= 1) acc += __shfl_xor(acc, off, 32);
  if (lane == 0) invn[wid] = 1.0f / fmaxf(sqrtf(acc), 1e-12f);
}

// ---------------- per-node state init ----------------
__global__ __launch_bounds__(256) void init_nodes_kernel(
    unsigned long long* __restrict__ k1, unsigned long long* __restrict__ k2,
    float* __restrict__ deg, int N, int zeroDeg) {
  int n = blockIdx.x * blockDim.x + threadIdx.x;
  if (n >= N) return;
  k1[n] = 0ull;
  k2[n] = 0ull;
  if (zeroDeg) deg[n] = 0.0f;
}

// ---------------- edge pass 1: cosine sim + argmax key + degree ----------------
// one wave per edge; each lane does one b128/b64 load per endpoint
template <int C>
__global__ __launch_bounds__(256) void edge_pass1_kernel(
    const float* __restrict__ H, const float* __restrict__ invn,
    const int* __restrict__ src, const int* __restrict__ dst,
    float* __restrict__ we, unsigned long long* __restrict__ best1,
    float* __restrict__ deg, int E, int countDeg) {
  int wid = blockIdx.x * (blockDim.x >> 5) + (threadIdx.x >> 5);
  if (wid >= E) return;
  int lane = threadIdx.x & 31;
  int s = src[wid], d = dst[wid];
  float acc;
  if constexpr (C == 128) {
    float4 a = *(const float4*)(H + (size_t)s * C + lane * 4);
    float4 b = *(const float4*)(H + (size_t)d * C + lane * 4);
    acc = a.x * b.x + a.y * b.y + a.z * b.z + a.w * b.w;
  } else {
    float2 a = *(const float2*)(H + (size_t)s * C + lane * 2);
    float2 b = *(const float2*)(H + (size_t)d * C + lane * 2);
    acc = a.x * b.x + a.y * b.y;
  }
  for (int off = 16; off; off >>= 1) acc += __shfl_xor(acc, off, 32);
  float w = acc * invn[s] * invn[d];
  if (lane == 0) {
    we[wid] = w;
    unsigned long long key =
        ((unsigned long long)fkey_enc(w) << 32) | (unsigned)(~(unsigned)wid);
    atomicMax(&best1[d], key);
    if (countDeg) atomicAdd(&deg[d], 1.0f);
  }
}

// ---------------- edge pass 2: second-best excluding the winner ----------------
__global__ __launch_bounds__(256) void edge_pass2_kernel(
    const float* __restrict__ we, const int* __restrict__ dst,
    const unsigned long long* __restrict__ best1,
    unsigned long long* __restrict__ best2, int E) {
  int e = blockIdx.x * blockDim.x + threadIdx.x;
  if (e >= E) return;
  int d = dst[e];
  unsigned e1 = ~(unsigned)(best1[d] & 0xFFFFFFFFull);
  if ((unsigned)e == e1) return;
  unsigned long long key =
      ((unsigned long long)fkey_enc(we[e]) << 32) | (unsigned)(~(unsigned)e);
  atomicMax(&best2[d], key);
}

// ---------------- aggregation: out[n] = (w1*H[s1] + w2*H[s2]) / max(deg,1) ----------------
// float4 per thread -> b128 gathers/stores; threshold w >= 0.0 <=> key-hi >= 0x80000000
template <int C>
__global__ __launch_bounds__(256) void aggregate_kernel(
    const float* __restrict__ H, const int* __restrict__ src,
    const unsigned long long* __restrict__ best1,
    const unsigned long long* __restrict__ best2,
    const float* __restrict__ deg, float* __restrict__ out,
    int N, int doRelu) {
  constexpr int CV = C / 4;
  int idx = blockIdx.x * blockDim.x + threadIdx.x;   // over N * C/4 float4s
  if (idx >= N * CV) return;
  int n  = idx / CV;
  int c  = (idx - n * CV) * 4;
  float sx = 0.f, sy = 0.f, sz = 0.f, sw = 0.f;
  unsigned long long k1 = best1[n];
  unsigned h1 = (unsigned)(k1 >> 32);
  if (h1 >= 0x80000000u) {
    unsigned e1 = ~(unsigned)k1;
    float w1 = fkey_dec(h1);
    float4 v = *(const float4*)(H + (size_t)src[e1] * C + c);
    sx += w1 * v.x; sy += w1 * v.y; sz += w1 * v.z; sw += w1 * v.w;
  }
  unsigned long long k2 = best2[n];
  unsigned h2 = (unsigned)(k2 >> 32);
  if (h2 >= 0x80000000u) {
    unsigned e2 = ~(unsigned)k2;
    float w2 = fkey_dec(h2);
    float4 v = *(const float4*)(H + (size_t)src[e2] * C + c);
    sx += w2 * v.x; sy += w2 * v.y; sz += w2 * v.z; sw += w2 * v.w;
  }
  float inv = 1.0f / fmaxf(deg[n], 1.0f);
  float4 r;
  r.x = sx * inv; r.y = sy * inv; r.z = sz * inv; r.w = sw * inv;
  if (doRelu) {
    r.x = fmaxf(r.x, 0.f); r.y = fmaxf(r.y, 0.f);
    r.z = fmaxf(r.z, 0.f); r.w = fmaxf(r.w, 0.f);
  }
  *(float4*)(out + (size_t)n * C + c) = r;
}

// ---------------- log_softmax over 64 columns (one wave per node) ----------------
__global__ __launch_bounds__(256) void logsoftmax64_kernel(float* __restrict__ out, int N) {
  int wid = blockIdx.x * (blockDim.x >> 5) + (threadIdx.x >> 5);
  if (wid >= N) return;
  int lane = threadIdx.x & 31;
  long base = (long)wid * OUT_DIM;
  float2 v = *(const float2*)(out + base + lane * 2);
  float m = fmaxf(v.x, v.y);
  for (int off = 16; off; off >>= 1) m = fmaxf(m, __shfl_xor(m, off, 32));
  float s = expf(v.x - m) + expf(v.y - m);
  for (int off = 16; off; off >>= 1) s += __shfl_xor(s, off, 32);
  float lg = m + logf(s);
  float2 r; r.x = v.x - lg; r.y = v.y - lg;
  *(float2*)(out + base + lane * 2) = r;
}

// =====================================================================
extern "C" void kernel_launch(void* const* d_in, const int* in_sizes, int n_in,
                              void* d_out, int out_size, void* d_ws, size_t ws_size,
                              hipStream_t stream) {
  const float* x   = (const float*)d_in[0];
  const int*   ei  = (const int*)d_in[1];
  const float* W1  = (const float*)d_in[2];
  const float* b1  = (const float*)d_in[3];
  const float* W2  = (const float*)d_in[4];
  const float* b2  = (const float*)d_in[5];

  const int N = in_sizes[0] / IN_DIM;   // 100000
  const int E = in_sizes[1] / 2;        // 1600000
  const int* src = ei;
  const int* dst = ei + E;

  // ---- workspace carve-up (all chunks naturally 16B-aligned for these sizes)
  char* p = (char*)d_ws;
  float* h1 = (float*)p;  p += (size_t)N * HID_DIM * sizeof(float);
  float* a1 = (float*)p;  p += (size_t)N * HID_DIM * sizeof(float);
  float* invn = (float*)p; p += (size_t)N * sizeof(float);
  float* deg  = (float*)p; p += (size_t)N * sizeof(float);
  unsigned long long* k1 = (unsigned long long*)p; p += (size_t)N * 8;
  unsigned long long* k2 = (unsigned long long*)p; p += (size_t)N * 8;
  float* we = (float*)p;  p += (size_t)E * sizeof(float);
  float* h2 = h1;                        // layer-1 features dead after aggregate-1
  float* out = (float*)d_out;

  const int mtiles = (N + 15) / 16;
  const int nodeW  = (N + 7) / 8;        // wave-per-node grids (8 waves / 256-blk)
  const int edgeW  = (E + 7) / 8;        // wave-per-edge grids
  const int nodeT  = (N + 255) / 256;
  const int edgeT  = (E + 255) / 256;

  // ================= layer 1 (C = 128, ReLU) =================
  gemm_wmma_kernel<HID_DIM><<<mtiles, 256, 0, stream>>>(x, W1, b1, h1, N);
  rownorm_kernel<HID_DIM><<<nodeW, 256, 0, stream>>>(h1, invn, N);
  init_nodes_kernel<<<nodeT, 256, 0, stream>>>(k1, k2, deg, N, 1);
  edge_pass1_kernel<HID_DIM><<<edgeW, 256, 0, stream>>>(h1, invn, src, dst, we,
                                                        k1, deg, E, 1);
  edge_pass2_kernel<<<edgeT, 256, 0, stream>>>(we, dst, k1, k2, E);
  {
    int tot = N * (HID_DIM / 4);
    aggregate_kernel<HID_DIM><<<(tot + 255) / 256, 256, 0, stream>>>(
        h1, src, k1, k2, deg, a1, N, 1);
  }

  // ================= layer 2 (C = 64, no ReLU) =================
  gemm_wmma_kernel<OUT_DIM><<<mtiles, 128, 0, stream>>>(a1, W2, b2, h2, N);
  rownorm_kernel<OUT_DIM><<<nodeW, 256, 0, stream>>>(h2, invn, N);
  init_nodes_kernel<<<nodeT, 256, 0, stream>>>(k1, k2, deg, N, 0);
  edge_pass1_kernel<OUT_DIM><<<edgeW, 256, 0, stream>>>(h2, invn, src, dst, we,
                                                        k1, deg, E, 0);
  edge_pass2_kernel<<<edgeT, 256, 0, stream>>>(we, dst, k1, k2, E);
  {
    int tot = N * (OUT_DIM / 4);
    aggregate_kernel<OUT_DIM><<<(tot + 255) / 256, 256, 0, stream>>>(
        h2, src, k1, k2, deg, out, N, 0);
  }
  logsoftmax64_kernel<<<nodeW, 256, 0, stream>>>(out, N);
}